// Feature_Net_83940840833059
// MI455X (gfx1250) — compile-verified
//
#include <hip/hip_runtime.h>

#define N_NODES  50000
#define N_EDGES  800000
#define NF       128
#define N_GRAPHS 128

typedef __attribute__((ext_vector_type(2))) float v2f;
typedef __attribute__((ext_vector_type(8))) float v8f;

// ---------------------------------------------------------------- utilities
__global__ void zero_kernel(float* __restrict__ p, int n) {
  int i = blockIdx.x * blockDim.x + threadIdx.x;
  if (i < n) p[i] = 0.0f;
}

// degree: one thread per edge, count into deg[dst]
__global__ void deg_kernel(const int* __restrict__ dst, float* __restrict__ deg) {
  int e = blockIdx.x * blockDim.x + threadIdx.x;
  if (e < N_EDGES) atomicAdd(&deg[dst[e]], 1.0f);
}

// dinv = rsqrt(deg + 1)   (self-loop included)
__global__ void dinv_kernel(float* __restrict__ deg) {
  int i = blockIdx.x * blockDim.x + threadIdx.x;
  if (i < N_NODES) deg[i] = rsqrtf(deg[i] + 1.0f);
}

// ---------------------------------------------------------------- WMMA GEMM
// H[N_NODES x 128] = X[N_NODES x 128] @ W[128 x 128]
// block = 256 threads = 8 waves; block covers 16 rows x 128 cols;
// each wave owns one 16x16 tile (cols = wave*16). 50000 = 3125 * 16 exactly,
// so EXEC is all-1s in every wave (WMMA requirement).
__global__ __launch_bounds__(256) void gemm_wmma(const float* __restrict__ X,
                                                 const float* __restrict__ W,
                                                 float* __restrict__ H) {
  const int wave = threadIdx.x >> 5;
  const int lane = threadIdx.x & 31;
  const int half = lane >> 4;   // 0: lanes 0-15, 1: lanes 16-31
  const int l    = lane & 15;
  const int m0   = blockIdx.x * 16;
  const int n0   = wave * 16;

  const float* Xrow = X + (size_t)(m0 + l) * NF;   // A: row M = m0 + l
  v8f acc = {};

  #pragma unroll
  for (int k = 0; k < NF; k += 4) {
    // A 16x4: lanes 0-15 hold K = k..k+1, lanes 16-31 hold K = k+2..k+3
    v2f a = *(const v2f*)(Xrow + k + 2 * half);           // 8B aligned
    // B 4x16: same K split, N = n0 + l
    v2f b;
    b.x = W[(size_t)(k + 2 * half + 0) * NF + n0 + l];
    b.y = W[(size_t)(k + 2 * half + 1) * NF + n0 + l];
    acc = __builtin_amdgcn_wmma_f32_16x16x4_f32(
        /*neg_a=*/false, a, /*neg_b=*/false, b,
        /*c_mod=*/(short)0, acc, /*reuse_a=*/false, /*reuse_b=*/false);
  }

  // C/D layout: VGPR r holds row M = m0 + r + 8*half, col N = n0 + l
  float* Hp = H + (size_t)(m0 + 8 * half) * NF + n0 + l;
  #pragma unroll
  for (int r = 0; r < 8; ++r) Hp[(size_t)r * NF] = acc[r];
}

// ---------------------------------------------------------------- scatter
// one wave per edge; lane handles 4 consecutive features (float4 gather +
// 4 coalesced f32 atomics). AGG[dst] += dinv[src]*dinv[dst] * H[src]
__global__ __launch_bounds__(256) void scatter_kernel(const float* __restrict__ H,
                                                      const int* __restrict__ src,
                                                      const int* __restrict__ dst,
                                                      const float* __restrict__ dinv,
                                                      float* __restrict__ AGG) {
  int gid  = blockIdx.x * blockDim.x + threadIdx.x;
  int e    = gid >> 5;
  if (e >= N_EDGES) return;
  int lane = gid & 31;
  int s = src[e];
  int d = dst[e];
  float norm = dinv[s] * dinv[d];
  const float4 hv = *(const float4*)(H + (size_t)s * NF + lane * 4);
  float* op = AGG + (size_t)d * NF + lane * 4;
  atomicAdd(op + 0, hv.x * norm);
  atomicAdd(op + 1, hv.y * norm);
  atomicAdd(op + 2, hv.z * norm);
  atomicAdd(op + 3, hv.w * norm);
}

// ---------------------------------------------------------------- combines
// Y = relu(AGG + dinv^2 * H + b)   (layer 1)
__global__ __launch_bounds__(256) void combine_relu(const float* __restrict__ AGG,
                                                    const float* __restrict__ H,
                                                    const float* __restrict__ dinv,
                                                    const float* __restrict__ b,
                                                    float* __restrict__ Y) {
  int gid = blockIdx.x * blockDim.x + threadIdx.x;   // over N_NODES*32
  if (gid >= N_NODES * 32) return;
  int i = gid >> 5;
  int q = (gid & 31) * 4;
  float di = dinv[i];
  float di2 = di * di;
  size_t off = (size_t)i * NF + q;
  float4 av = *(const float4*)(AGG + off);
  float4 hv = *(const float4*)(H + off);
  float4 bv = *(const float4*)(b + q);
  float4 r;
  r.x = fmaxf(av.x + di2 * hv.x + bv.x, 0.0f);
  r.y = fmaxf(av.y + di2 * hv.y + bv.y, 0.0f);
  r.z = fmaxf(av.z + di2 * hv.z + bv.z, 0.0f);
  r.w = fmaxf(av.w + di2 * hv.w + bv.w, 0.0f);
  *(float4*)(Y + off) = r;
}

// layer 2 combine fused with global_add_pool: out[batch[i]] += val
__global__ __launch_bounds__(256) void combine_pool(const float* __restrict__ AGG,
                                                    const float* __restrict__ H,
                                                    const float* __restrict__ dinv,
                                                    const float* __restrict__ b,
                                                    const int* __restrict__ batch,
                                                    float* __restrict__ out) {
  int gid = blockIdx.x * blockDim.x + threadIdx.x;   // over N_NODES*32
  if (gid >= N_NODES * 32) return;
  int i = gid >> 5;
  int q = (gid & 31) * 4;
  float di = dinv[i];
  float di2 = di * di;
  size_t off = (size_t)i * NF + q;
  float4 av = *(const float4*)(AGG + off);
  float4 hv = *(const float4*)(H + off);
  float4 bv = *(const float4*)(b + q);
  float* op = out + (size_t)batch[i] * NF + q;
  atomicAdd(op + 0, av.x + di2 * hv.x + bv.x);
  atomicAdd(op + 1, av.y + di2 * hv.y + bv.y);
  atomicAdd(op + 2, av.z + di2 * hv.z + bv.z);
  atomicAdd(op + 3, av.w + di2 * hv.w + bv.w);
}

// ---------------------------------------------------------------- launch
extern "C" void kernel_launch(void* const* d_in, const int* in_sizes, int n_in,
                              void* d_out, int out_size, void* d_ws, size_t ws_size,
                              hipStream_t stream) {
  const float* x    = (const float*)d_in[0];
  const int*   ei   = (const int*)  d_in[1];
  const int*   batch= (const int*)  d_in[2];
  const float* W1   = (const float*)d_in[3];
  const float* b1   = (const float*)d_in[4];
  const float* W2   = (const float*)d_in[5];
  const float* b2   = (const float*)d_in[6];
  float* out = (float*)d_out;
  float* ws  = (float*)d_ws;

  const int* src = ei;             // edge_index[0]
  const int* dst = ei + N_EDGES;   // edge_index[1]

  float* dinv = ws;                         // 50000 floats
  float* H    = ws + 50048;                 // N_NODES*NF
  float* AGG  = H  + (size_t)N_NODES * NF;  // N_NODES*NF
  float* Y    = AGG + (size_t)N_NODES * NF; // N_NODES*NF

  const int T = 256;
  const int gNodes    = (N_NODES + T - 1) / T;
  const int gEdges    = (N_EDGES + T - 1) / T;
  const int gFeat     = ((size_t)N_NODES * NF + T - 1) / T;       // zero agg
  const int gNodeVec  = (N_NODES * 32 + T - 1) / T;               // combines
  const int gScatter  = ((size_t)N_EDGES * 32 + T - 1) / T;
  const int gGemm     = N_NODES / 16;                             // 3125

  // degrees -> dinv
  zero_kernel<<<gNodes, T, 0, stream>>>(dinv, N_NODES);
  deg_kernel<<<gEdges, T, 0, stream>>>(dst, dinv);
  dinv_kernel<<<gNodes, T, 0, stream>>>(dinv);

  // ---- layer 1
  gemm_wmma<<<gGemm, T, 0, stream>>>(x, W1, H);
  zero_kernel<<<gFeat, T, 0, stream>>>(AGG, N_NODES * NF);
  scatter_kernel<<<gScatter, T, 0, stream>>>(H, src, dst, dinv, AGG);
  combine_relu<<<gNodeVec, T, 0, stream>>>(AGG, H, dinv, b1, Y);

  // ---- layer 2
  gemm_wmma<<<gGemm, T, 0, stream>>>(Y, W2, H);
  zero_kernel<<<gFeat, T, 0, stream>>>(AGG, N_NODES * NF);
  scatter_kernel<<<gScatter, T, 0, stream>>>(H, src, dst, dinv, AGG);

  // ---- combine + pool
  zero_kernel<<<(N_GRAPHS * NF + T - 1) / T, T, 0, stream>>>(out, N_GRAPHS * NF);
  combine_pool<<<gNodeVec, T, 0, stream>>>(AGG, H, dinv, b2, batch, out);
}